// NumDualDescriptorPM_48661979463961
// MI455X (gfx1250) — compile-verified
//
#include <hip/hip_runtime.h>
#include <math.h>

// ---------------------------------------------------------------------------
// NumDualDescriptorPM for MI455X (gfx1250, wave32)
//
//   x  = vectors @ M_weight.T                      (2048x256x256 fp32 GEMM)
//   Nk[b,i] = sum_j x[b,j] * P[i,j] * cos(2*pi*k_b / (i*256 + j + 2))
//
// GEMM -> V_WMMA_F32_16X16X4_F32 (full fp32, matches reference precision).
// Einsum -> VALU/TRANS bound (134M cos, no matmul factorization exists since
// the phase couples b,i,j). LDS-tiled with padding to avoid bank conflicts.
// ---------------------------------------------------------------------------

typedef __attribute__((ext_vector_type(2))) float v2f;
typedef __attribute__((ext_vector_type(8))) float v8f;

#define VEC_DIM 256
#define BATCH   2048
#define TWO_PI  6.28318530717958647692f

// ---------------------------------------------------------------------------
// Kernel A: x[b,n] = sum_k vectors[b,k] * M_weight[n,k]
// One wave computes one 16x16 output tile; K advances 4 per WMMA.
// ISA operand layouts (CDNA5 ISA, 32-bit A 16x4 & C/D 16x16):
//   A: lane l<16  -> (M=l,    K=k+0..k+1) in v[0..1]
//      lane l>=16 -> (M=l-16, K=k+2..k+3) in v[0..1]   => contiguous float2
//   B: mirrored with lane = N column, same K split     => contiguous float2
//   C/D: VGPR r, lanes 0-15 -> M=r, lanes 16-31 -> M=r+8, N=lane&15
// ---------------------------------------------------------------------------
__global__ __launch_bounds__(128) void gemm_wmma_f32_k(
    const float* __restrict__ V,   // [BATCH, VEC_DIM]
    const float* __restrict__ W,   // [VEC_DIM, VEC_DIM] (M_weight)
    float* __restrict__ X)         // [BATCH, VEC_DIM]
{
    const int lane  = threadIdx.x & 31;
    const int wave  = threadIdx.x >> 5;
    const int tile  = blockIdx.x * 4 + wave;          // 4 waves per block
    const int nT    = VEC_DIM / 16;                   // 16 tiles along N
    const int m0    = (tile / nT) * 16;               // batch-row base
    const int n0    = (tile % nT) * 16;               // out-col base

    const int lm    = lane & 15;
    const int khalf = (lane >> 4) * 2;                // 0 or 2

    const float* aPtr = V + (size_t)(m0 + lm) * VEC_DIM + khalf; // A row = batch
    const float* bPtr = W + (size_t)(n0 + lm) * VEC_DIM + khalf; // B col n = W row n

    v8f c = {0.f, 0.f, 0.f, 0.f, 0.f, 0.f, 0.f, 0.f};

    #pragma unroll 4
    for (int k = 0; k < VEC_DIM; k += 4) {
        v2f a = *(const v2f*)(aPtr + k);              // global_load_b64
        v2f b = *(const v2f*)(bPtr + k);              // global_load_b64
        // (neg_a, A, neg_b, B, c_mod, C, reuse_a, reuse_b)
        c = __builtin_amdgcn_wmma_f32_16x16x4_f32(
                false, a, false, b, (short)0, c, false, false);
    }

    const int rowAdd = (lane >> 4) * 8;               // lanes 16-31 -> M+8
    float* out = X + (size_t)(m0 + rowAdd) * VEC_DIM + (n0 + lm);
    #pragma unroll
    for (int r = 0; r < 8; ++r)
        out[(size_t)r * VEC_DIM] = c[r];
}

// ---------------------------------------------------------------------------
// Kernel B: Nk[b,i] = sum_j x[b,j] * P[i,j] * cos(TWO_PI*k_b / (i*256+j+2))
// Block = 256 threads = one 16(b) x 16(i) output tile, one output per thread.
// x rows and P rows staged in LDS with stride 257: P access ps[ii][j] would be
// a 16-way bank conflict at stride 256 (256 % 64 == 0); 257 % 64 == 1 fixes it.
// Period i*256+j+2 <= 65537 < 2^24 -> exact in fp32; angle formed exactly as
// the reference (fp32 mul/div), cosf does accurate range reduction.
// ---------------------------------------------------------------------------
__global__ __launch_bounds__(256) void einsum_cos_k(
    const float* __restrict__ K,   // [BATCH] k_tensor
    const float* __restrict__ X,   // [BATCH, VEC_DIM] (from kernel A)
    const float* __restrict__ P,   // [VEC_DIM, VEC_DIM]
    float* __restrict__ Nk)        // [BATCH, VEC_DIM]
{
    __shared__ float xs[16][257];
    __shared__ float ps[16][257];

    const int t  = threadIdx.x;
    const int b0 = blockIdx.x * 16;
    const int i0 = blockIdx.y * 16;

    // Coalesced cooperative loads: 16 rows x 256 cols each.
    for (int idx = t; idx < 16 * VEC_DIM; idx += 256) {
        const int r = idx >> 8;
        const int cch = idx & 255;
        xs[r][cch] = X[(size_t)(b0 + r) * VEC_DIM + cch];
        ps[r][cch] = P[(size_t)(i0 + r) * VEC_DIM + cch];
    }
    __syncthreads();

    const int bi = t >> 4;                            // local batch   0..15
    const int ii = t & 15;                            // local i-row   0..15
    const int ig = i0 + ii;

    const float kb    = K[b0 + bi];
    const float num   = TWO_PI * kb;
    const float pbase = (float)(ig * VEC_DIM + 2);    // exact (< 2^24)

    float acc = 0.0f;
    #pragma unroll 4
    for (int j = 0; j < VEC_DIM; ++j) {
        const float ang = num / (pbase + (float)j);
        acc = fmaf(xs[bi][j] * ps[ii][j], cosf(ang), acc);
    }

    Nk[(size_t)(b0 + bi) * VEC_DIM + ig] = acc;
}

// ---------------------------------------------------------------------------
extern "C" void kernel_launch(void* const* d_in, const int* in_sizes, int n_in,
                              void* d_out, int out_size, void* d_ws, size_t ws_size,
                              hipStream_t stream)
{
    const float* k_tensor = (const float*)d_in[0];   // [2048]
    const float* vectors  = (const float*)d_in[1];   // [2048, 256]
    const float* M_weight = (const float*)d_in[2];   // [256, 256]
    const float* P        = (const float*)d_in[3];   // [256, 256]
    float*       out      = (float*)d_out;           // [2048, 256]
    float*       X        = (float*)d_ws;            // scratch: 2048*256*4 = 2 MB

    (void)in_sizes; (void)n_in; (void)out_size; (void)ws_size;

    // Kernel A: WMMA GEMM. 2048 16x16 tiles, 4 waves (128 thr) per block.
    const int tiles = (BATCH / 16) * (VEC_DIM / 16); // 2048
    gemm_wmma_f32_k<<<tiles / 4, 128, 0, stream>>>(vectors, M_weight, X);

    // Kernel B: cos-weighted contraction (reads X — stream-ordered after A).
    dim3 grid(BATCH / 16, VEC_DIM / 16);             // 128 x 16 blocks
    einsum_cos_k<<<grid, 256, 0, stream>>>(k_tensor, X, P, out);
}